// GINLayer_39771397161473
// MI455X (gfx1250) — compile-verified
//
#include <hip/hip_runtime.h>

// ---------------------------------------------------------------------------
// GIN layer for MI455X (gfx1250, wave32):
//   1) X_agg = X ; scatter-add both edge directions (f32 atomics, L2-resident)
//   2) out   = relu(X_agg @ (W_h @ W_o) + (b_h @ W_o + b_o))
//      using V_WMMA_F32_16X16X4_F32 (exact fp32 matrix path)
// X_agg lives in d_out (same shape); GEMM runs in place per 16-row tile.
// ---------------------------------------------------------------------------

#define N_NODES 100000
#define N_EDGES 640000
#define D_FEAT  128

typedef __attribute__((ext_vector_type(2))) float v2f;
typedef __attribute__((ext_vector_type(8))) float v8f;

// --- combine the two dense layers: Wc = Wh @ Wo ; bc = bh @ Wo + bo ---------
__global__ void gin_prep_weights(const float* __restrict__ Wh,
                                 const float* __restrict__ bh,
                                 const float* __restrict__ Wo,
                                 const float* __restrict__ bo,
                                 float* __restrict__ Wc,
                                 float* __restrict__ bc) {
    const int j = threadIdx.x;      // output column 0..127
    const int i = blockIdx.x;       // row 0..127, block 128 = bias
    if (i < D_FEAT) {
        float s = 0.f;
        #pragma unroll 8
        for (int k = 0; k < D_FEAT; ++k)
            s = fmaf(Wh[i * D_FEAT + k], Wo[k * D_FEAT + j], s);
        Wc[i * D_FEAT + j] = s;
    } else {
        float s = bo[j];
        #pragma unroll 8
        for (int k = 0; k < D_FEAT; ++k)
            s = fmaf(bh[k], Wo[k * D_FEAT + j], s);
        bc[j] = s;
    }
}

// --- X_agg = X (float4 streaming copy) --------------------------------------
__global__ void gin_copy(const float4* __restrict__ X, float4* __restrict__ Xagg) {
    const int i = blockIdx.x * blockDim.x + threadIdx.x;   // grid sized exactly
    Xagg[i] = X[i];
}

// --- undirected scatter-add: thread = (edge, group-of-4-features) -----------
__global__ void gin_scatter(const float4* __restrict__ X,
                            const int* __restrict__ ra,
                            const int* __restrict__ rb,
                            float* __restrict__ Xagg) {
    const unsigned t = blockIdx.x * blockDim.x + threadIdx.x;
    const int e = (int)(t >> 5);           // edge index
    const int g = (int)(t & 31);           // float4 group (32 * 4 = 128 feats)
    if (e >= N_EDGES) return;
    const int a = ra[e];
    const int b = rb[e];
    // gathers hit L2 (X = 51 MB < 192 MB L2)
    const float4 va = X[a * (D_FEAT / 4) + g];
    const float4 vb = X[b * (D_FEAT / 4) + g];
    float* pb = Xagg + (size_t)b * D_FEAT + g * 4;
    float* pa = Xagg + (size_t)a * D_FEAT + g * 4;
    atomicAdd(pb + 0, va.x); atomicAdd(pb + 1, va.y);
    atomicAdd(pb + 2, va.z); atomicAdd(pb + 3, va.w);
    atomicAdd(pa + 0, vb.x); atomicAdd(pa + 1, vb.y);
    atomicAdd(pa + 2, vb.z); atomicAdd(pa + 3, vb.w);
}

// --- fused GEMM + bias + relu, in place on X_agg (= d_out) ------------------
// Block = 256 threads = 8 waves; block owns 16 rows; wave w owns columns
// [16w, 16w+16). One V_WMMA_F32_16X16X4_F32 per K-step of 4.
__global__ void gin_gemm(float* __restrict__ Xagg,
                         const float* __restrict__ Wc,
                         const float* __restrict__ bc) {
    const int lane = threadIdx.x & 31;
    const int wave = threadIdx.x >> 5;
    const int m    = lane & 15;        // A row / B-C-D column slot
    const int h    = lane >> 4;        // half-wave select
    const int n    = (wave << 4) + m;  // output column for this lane
    const size_t row = (size_t)blockIdx.x * 16 + m;
    const float* __restrict__ arow = Xagg + row * D_FEAT;

    v8f acc = {};
    #pragma unroll 4
    for (int k0 = 0; k0 < D_FEAT; k0 += 4) {
        const int k = k0 + 2 * h;
        v2f a;                               // A[m][k], A[m][k+1]
        a.x = arow[k];
        a.y = arow[k + 1];
        v2f b;                               // B[k][n], B[k+1][n]
        b.x = Wc[k * D_FEAT + n];
        b.y = Wc[(k + 1) * D_FEAT + n];
        acc = __builtin_amdgcn_wmma_f32_16x16x4_f32(
            /*neg_a=*/false, a, /*neg_b=*/false, b,
            /*c_mod=*/(short)0, acc, /*reuse_a=*/false, /*reuse_b=*/false);
    }

    const float bias = bc[n];
    // All A-reads of this block's 16 rows are consumed above; barrier before
    // overwriting those rows in place.
    __syncthreads();

    const size_t base = (size_t)blockIdx.x * 16;
    #pragma unroll
    for (int r = 0; r < 8; ++r) {
        float v = acc[r] + bias;             // D row = r + 8h, col = n
        v = v > 0.f ? v : 0.f;
        Xagg[(base + r + 8 * h) * D_FEAT + n] = v;
    }
}

extern "C" void kernel_launch(void* const* d_in, const int* in_sizes, int n_in,
                              void* d_out, int out_size, void* d_ws, size_t ws_size,
                              hipStream_t stream) {
    const float* X  = (const float*)d_in[0];   // [N, 128]
    const float* Wh = (const float*)d_in[1];   // [128, 128]
    const float* bh = (const float*)d_in[2];   // [128]
    const float* Wo = (const float*)d_in[3];   // [128, 128]
    const float* bo = (const float*)d_in[4];   // [128]
    const int*   ra = (const int*)d_in[5];     // [E]
    const int*   rb = (const int*)d_in[6];     // [E]

    float* out = (float*)d_out;                // doubles as X_agg
    float* Wc  = (float*)d_ws;                 // 128*128 combined weights
    float* bc  = Wc + D_FEAT * D_FEAT;         // 128 combined bias

    // 1) combine weights (tiny)
    gin_prep_weights<<<D_FEAT + 1, D_FEAT, 0, stream>>>(Wh, bh, Wo, bo, Wc, bc);

    // 2) X_agg = X   (100000*128/4 = 3,200,000 float4 / 256 = 12500 blocks)
    gin_copy<<<(N_NODES * (D_FEAT / 4)) / 256, 256, 0, stream>>>(
        (const float4*)X, (float4*)out);

    // 3) scatter-add both directions: 640000 edges * 32 groups / 256 = 80000
    gin_scatter<<<(N_EDGES * 32) / 256, 256, 0, stream>>>(
        (const float4*)X, ra, rb, out);

    // 4) fused GEMM + bias + relu in place (100000 / 16 = 6250 tiles)
    gin_gemm<<<N_NODES / 16, 256, 0, stream>>>(out, Wc, bc);
}